// SNN_MLP_39840116637822
// MI455X (gfx1250) — compile-verified
//
#include <hip/hip_runtime.h>
#include <hip/hip_bf16.h>
#include <math.h>

// ---------------------------------------------------------------------------
// SNN MLP step: B=128, IN=784, H=400, OUT=10, fp32.
// Layer-1 event scan done as blocked lower-triangular matmuls on the fp32
// matrix pipe (V_WMMA_F32_16X16X4_F32), per-neuron masked running max/min,
// one cross-lane reduction at the end. Layer 2 + finalize in a per-batch wave.
// ---------------------------------------------------------------------------

typedef float v2f __attribute__((ext_vector_type(2)));
typedef float v8f __attribute__((ext_vector_type(8)));

#define B_      128
#define IN_     784
#define H_      400
#define OUT_    10
#define NTILES  25      // H_/16
#define ETILES  49      // IN_/16
#define THRESH_ 0.5f
#define DECAY_  0.2f

// flat d_out offsets (return order: h2_spiked, step_max, step_min, h1_mem, h2_mem)
#define OFF_H2S 0
#define OFF_MAX 1280
#define OFF_MIN 1408
#define OFF_H1M 1536
#define OFF_H2M 52736

// ------------------------- layer 1: WMMA prefix scan -----------------------
// grid(25, 128), one wave32 per block. Block handles one (neuron-tile, batch).
__global__ __launch_bounds__(32)
void snn_l1_wmma_scan(const float* __restrict__ input,
                      const float* __restrict__ W1,
                      const float* __restrict__ h1m0,
                      float* __restrict__ h1_final,   // raw finals -> rewritten by kernel 2
                      float* __restrict__ ws_max,     // [B, NTILES]
                      float* __restrict__ ws_min)     // [B, NTILES]
{
    __shared__ float sgate[IN_];
    const int ntile = blockIdx.x;
    const int b     = blockIdx.y;
    const int lane  = threadIdx.x;
    const int lid   = lane & 15;     // column n / A-row m
    const int hi    = lane >> 4;     // lane half selects K / row groups

    // stage this batch's gate row (784 floats) in LDS
    const float* grow = input + b * IN_;
    for (int i = lane; i < IN_; i += 32) sgate[i] = grow[i];
    __syncthreads();

    const int    n_g  = ntile * 16 + lid;
    const float* wrow = W1 + n_g * IN_;          // B[e',n] = W1[n, e']

    float carry  = h1m0[n_g];                    // running membrane for column n_g
    float runMax = -__builtin_inff();
    float runMin =  __builtin_inff();
    const int m  = lid;                          // A-matrix row (event index in tile)

    for (int t = 0; t < ETILES; ++t) {
        const int tbase = t * 16;
        v8f acc = {};                            // S[e,n] intra-tile prefix sums
        #pragma unroll
        for (int c = 0; c < 4; ++c) {            // 4 chained K=4 fp32 WMMAs => K=16
            const int kA = c * 4 + hi * 2;       // K indices {kA, kA+1} for this lane
            v2f a, bm;
            // A[m, k] = (k <= m) ? gate[tbase+k] : 0   (masked lower-triangular)
            a.x = (kA + 0 <= m) ? sgate[tbase + kA + 0] : 0.0f;
            a.y = (kA + 1 <= m) ? sgate[tbase + kA + 1] : 0.0f;
            // B[k, n] = W1[n, tbase+k]
            bm.x = wrow[tbase + kA + 0];
            bm.y = wrow[tbase + kA + 1];
            acc = __builtin_amdgcn_wmma_f32_16x16x4_f32(
                /*neg_a=*/false, a, /*neg_b=*/false, bm,
                /*c_mod=*/(short)0, acc, /*reuse_a=*/false, /*reuse_b=*/false);
        }

        // tile column-sum = S[15, n]; it lives in acc[7] of lanes 16..31
        const float s15 = __shfl(acc[7], lid + 16, 32);

        // masked running max/min over the 8 event rows this lane owns
        #pragma unroll
        for (int r = 0; r < 8; ++r) {
            const int   e   = tbase + r + hi * 8;
            const float cum = carry + acc[r];    // snapshot after event e
            if (sgate[e] != 0.0f) {
                runMax = fmaxf(runMax, cum);
                runMin = fminf(runMin, cum);
            }
        }
        carry += s15;                            // advance running membrane
    }

    // wave-wide max/min over 16 neurons x both row halves
    #pragma unroll
    for (int off = 16; off >= 1; off >>= 1) {
        runMax = fmaxf(runMax, __shfl_xor(runMax, off, 32));
        runMin = fminf(runMin, __shfl_xor(runMin, off, 32));
    }
    if (lane == 0) {
        ws_max[b * NTILES + ntile] = runMax;
        ws_min[b * NTILES + ntile] = runMin;
    }
    if (hi == 0) h1_final[b * H_ + n_g] = carry; // lanes 0..15 carry the column finals
}

// ------------------- layer 2 scan + finalize (per batch) -------------------
// grid(128), one wave32 per block.
__global__ __launch_bounds__(32)
void snn_l2_finalize(const float* __restrict__ W2,
                     const float* __restrict__ h2m0,
                     float* __restrict__ out,
                     const float* __restrict__ ws_max,
                     const float* __restrict__ ws_min)
{
    __shared__ float sf[H_];
    const int b    = blockIdx.x;
    const int lane = threadIdx.x;
    float* h1reg = out + OFF_H1M + b * H_;       // raw h1 finals staged here

    for (int i = lane; i < H_; i += 32) sf[i] = h1reg[i];
    __syncthreads();

    // rewrite h1_mem in place: where spiked -> 0, else final * DECAY
    for (int i = lane; i < H_; i += 32) {
        const float f = sf[i];
        h1reg[i] = (f > THRESH_) ? 0.0f : f * DECAY_;
    }

    // layer-2 event scan: lane o < 10 owns one output neuron
    const bool valid = lane < OUT_;
    const int  o     = valid ? lane : 0;
    const float* w2row = W2 + o * H_;
    float m  = valid ? h2m0[o] : 0.0f;
    float rm = -__builtin_inff();
    float rn =  __builtin_inff();

    for (int e = 0; e < H_; ++e) {               // gate uniform across lanes
        if (sf[e] > THRESH_) {                   // h1_spiked[b,e] == 1
            m += w2row[e];
            if (valid) { rm = fmaxf(rm, m); rn = fminf(rn, m); }
        }
    }

    const float sp = (m > THRESH_) ? 1.0f : 0.0f;
    if (valid) {
        out[OFF_H2S + b * OUT_ + o] = sp;
        out[OFF_H2M + b * OUT_ + o] = (sp != 0.0f) ? 0.0f : m * DECAY_;
    }

    // reduce max2/min2 across lanes (invalid lanes hold +-inf identities)
    #pragma unroll
    for (int off = 16; off >= 1; off >>= 1) {
        rm = fmaxf(rm, __shfl_xor(rm, off, 32));
        rn = fminf(rn, __shfl_xor(rn, off, 32));
    }
    // fold in layer-1 per-tile partials
    float pm = (lane < NTILES) ? ws_max[b * NTILES + lane] : -__builtin_inff();
    float pn = (lane < NTILES) ? ws_min[b * NTILES + lane] :  __builtin_inff();
    #pragma unroll
    for (int off = 16; off >= 1; off >>= 1) {
        pm = fmaxf(pm, __shfl_xor(pm, off, 32));
        pn = fminf(pn, __shfl_xor(pn, off, 32));
    }
    if (lane == 0) {
        out[OFF_MAX + b] = fmaxf(rm, pm);
        out[OFF_MIN + b] = fminf(rn, pn);
    }
}

// --------------------------------- launch ----------------------------------
extern "C" void kernel_launch(void* const* d_in, const int* in_sizes, int n_in,
                              void* d_out, int out_size, void* d_ws, size_t ws_size,
                              hipStream_t stream) {
    const float* input = (const float*)d_in[0];  // [128, 784]
    const float* W1    = (const float*)d_in[1];  // [400, 784]
    const float* W2    = (const float*)d_in[2];  // [10, 400]
    const float* h1m0  = (const float*)d_in[3];  // [400]
    const float* h2m0  = (const float*)d_in[4];  // [10]
    float* out = (float*)d_out;

    float* ws_max = (float*)d_ws;                // 128*25 floats
    float* ws_min = ws_max + B_ * NTILES;        // 128*25 floats (25.6 KB total)

    snn_l1_wmma_scan<<<dim3(NTILES, B_), dim3(32), 0, stream>>>(
        input, W1, h1m0, out + OFF_H1M, ws_max, ws_min);
    snn_l2_finalize<<<dim3(B_), dim3(32), 0, stream>>>(
        W2, h2m0, out, ws_max, ws_min);
}